// GraphConvolutionBlock_43087111913571
// MI455X (gfx1250) — compile-verified
//
#include <hip/hip_runtime.h>
#include <hip/hip_bf16.h>
#include <math.h>

typedef float v2f __attribute__((ext_vector_type(2)));
typedef float v8f __attribute__((ext_vector_type(8)));

#define NB      4
#define C_IN    64
#define LSEQ    16384
#define NNODES  (NB * LSEQ * C_IN / C_IN * 1)      // placeholder, real below
#define NNODE   65536                              // B*L after view(-1, c) => (4*64*16384)/64
#define NHEAD   3
#define C_OUT   64
#define HF      (NHEAD * C_OUT)                    // 192
#define NEDGE   1048576
#define ETOT    (NEDGE + NNODE)                    // edges + self loops = 1114112
#define NEG_SLOPE 0.2f
#define BN_EPS  1e-5f

// ---------------- float atomic max (sign trick, init = -inf) ----------------
__device__ __forceinline__ void atomicMaxF(float* addr, float v) {
    if (v >= 0.f) atomicMax((int*)addr, __float_as_int(v));
    else          atomicMin((unsigned int*)addr, (unsigned int)__float_as_int(v));
}

// ---------------- K0: init accumulators ----------------
__global__ void k_init(float* __restrict__ m, float* __restrict__ denom,
                       float* __restrict__ outacc, float* __restrict__ stats) {
    int idx = blockIdx.x * blockDim.x + threadIdx.x;
    if (idx < NNODE * NHEAD) { m[idx] = __int_as_float(0xFF800000); denom[idx] = 0.f; }
    if (idx < 128) stats[idx] = 0.f;
    if (idx < NNODE * C_OUT) outacc[idx] = 0.f;
}

// ---------------- K1: h[N,192] = xf[N,64] @ W[192,64]^T  (WMMA f32 16x16x4) --
__global__ void k_gemm_h(const float* __restrict__ xf, const float* __restrict__ W,
                         float* __restrict__ h) {
    const int tilesN = HF / 16;                         // 12
    int wave = blockIdx.x * (blockDim.x >> 5) + (threadIdx.x >> 5);
    int lane = threadIdx.x & 31;
    int tm = wave / tilesN, tn = wave % tilesN;
    int m0 = tm * 16, n0 = tn * 16;
    int half = lane >> 4, l16 = lane & 15;
    const float* aptr = xf + (size_t)(m0 + l16) * C_IN + half * 2;  // A row, K half
    const float* bptr = W  + (size_t)(n0 + l16) * C_IN + half * 2;  // B col j, K half
    v8f c = {};
#pragma unroll
    for (int k0 = 0; k0 < C_IN; k0 += 4) {
        v2f a, b;
        a.x = aptr[k0 + 0]; a.y = aptr[k0 + 1];
        b.x = bptr[k0 + 0]; b.y = bptr[k0 + 1];
        c = __builtin_amdgcn_wmma_f32_16x16x4_f32(false, a, false, b, (short)0, c, false, false);
    }
#pragma unroll
    for (int r = 0; r < 8; ++r) {
        int row = m0 + half * 8 + r;
        h[(size_t)row * HF + n0 + l16] = c[r];
    }
}

// ---------------- K2: per-node attention dots ----------------
__global__ void k_att(const float* __restrict__ h, const float* __restrict__ att_src,
                      const float* __restrict__ att_dst,
                      float* __restrict__ a_s, float* __restrict__ a_d) {
    int idx = blockIdx.x * blockDim.x + threadIdx.x;       // n*3+hh
    if (idx >= NNODE * NHEAD) return;
    int n = idx / NHEAD, hh = idx % NHEAD;
    const float* hp = h + (size_t)n * HF + hh * C_OUT;
    const float* as = att_src + hh * C_OUT;
    const float* ad = att_dst + hh * C_OUT;
    float s = 0.f, d = 0.f;
#pragma unroll 8
    for (int f = 0; f < C_OUT; ++f) { float v = hp[f]; s += v * as[f]; d += v * ad[f]; }
    a_s[idx] = s; a_d[idx] = d;
}

// ---------------- K3: segment max over edges ----------------
__global__ void k_segmax(const int* __restrict__ ei, const float* __restrict__ a_s,
                         const float* __restrict__ a_d, float* __restrict__ m) {
    int idx = blockIdx.x * blockDim.x + threadIdx.x;       // e*3+hh
    if (idx >= ETOT * NHEAD) return;
    int e = idx / NHEAD, hh = idx % NHEAD;
    int s, d;
    if (e < NEDGE) { s = ei[e]; d = ei[NEDGE + e]; } else { s = d = e - NEDGE; }
    float v = a_s[s * NHEAD + hh] + a_d[d * NHEAD + hh];
    v = (v > 0.f) ? v : NEG_SLOPE * v;
    atomicMaxF(&m[d * NHEAD + hh], v);
}

// ---------------- K4: ex = exp(e - m[dst]); denom += ex ----------------
__global__ void k_expsum(const int* __restrict__ ei, const float* __restrict__ a_s,
                         const float* __restrict__ a_d, const float* __restrict__ m,
                         float* __restrict__ exbuf, float* __restrict__ denom) {
    int idx = blockIdx.x * blockDim.x + threadIdx.x;
    if (idx >= ETOT * NHEAD) return;
    int e = idx / NHEAD, hh = idx % NHEAD;
    int s, d;
    if (e < NEDGE) { s = ei[e]; d = ei[NEDGE + e]; } else { s = d = e - NEDGE; }
    float v = a_s[s * NHEAD + hh] + a_d[d * NHEAD + hh];
    v = (v > 0.f) ? v : NEG_SLOPE * v;
    float ex = expf(v - m[d * NHEAD + hh]);
    exbuf[idx] = ex;
    atomicAdd(&denom[d * NHEAD + hh], ex);
}

// ---------------- K5: wave-per-edge weighted scatter (head-mean fused) ------
__global__ void k_scatter(const int* __restrict__ ei, const float* __restrict__ h,
                          const float* __restrict__ exbuf, const float* __restrict__ denom,
                          float* __restrict__ outacc) {
    int wave = blockIdx.x * (blockDim.x >> 5) + (threadIdx.x >> 5);
    int lane = threadIdx.x & 31;
    if (wave >= ETOT) return;
    int s, d;
    if (wave < NEDGE) { s = ei[wave]; d = ei[NEDGE + wave]; } else { s = d = wave - NEDGE; }
    const float third = 1.f / 3.f;
    float al0 = exbuf[(size_t)wave * NHEAD + 0] / (denom[d * NHEAD + 0] + 1e-16f) * third;
    float al1 = exbuf[(size_t)wave * NHEAD + 1] / (denom[d * NHEAD + 1] + 1e-16f) * third;
    float al2 = exbuf[(size_t)wave * NHEAD + 2] / (denom[d * NHEAD + 2] + 1e-16f) * third;
    const float* hp = h + (size_t)s * HF;
    int f = lane;
    float acc0 = al0 * hp[f]      + al1 * hp[64 + f]  + al2 * hp[128 + f];
    float acc1 = al0 * hp[32 + f] + al1 * hp[96 + f]  + al2 * hp[160 + f];
    atomicAdd(&outacc[(size_t)d * C_OUT + f],        acc0);
    atomicAdd(&outacc[(size_t)d * C_OUT + 32 + f],   acc1);
}

// ---------------- K6: BN batch statistics (sum, sumsq per feature) ----------
__global__ void k_stats(const float* __restrict__ outacc, const float* __restrict__ gat_bias,
                        float* __restrict__ stats) {
    int f = threadIdx.x & 63;
    int rg = threadIdx.x >> 6;                              // 0..3 (blockDim=256)
    float gb = gat_bias[f];
    float s = 0.f, sq = 0.f;
    for (int n = blockIdx.x * 4 + rg; n < NNODE; n += gridDim.x * 4) {
        float v = outacc[(size_t)n * C_OUT + f] + gb;
        s += v; sq += v * v;
    }
    atomicAdd(&stats[f], s);
    atomicAdd(&stats[64 + f], sq);
}

// ---------------- K7: fold BN into per-feature scale/shift ----------------
__global__ void k_bnparams(const float* __restrict__ stats, const float* __restrict__ gamma,
                           const float* __restrict__ beta,
                           float* __restrict__ scale, float* __restrict__ shift) {
    int f = threadIdx.x;
    if (f >= C_OUT) return;
    const float invN = 1.f / (float)NNODE;
    float mu  = stats[f] * invN;
    float var = stats[64 + f] * invN - mu * mu;
    float sc  = gamma[f] * rsqrtf(var + BN_EPS);
    scale[f] = sc;
    shift[f] = beta[f] - mu * sc;
}

// ---------------- K8: y[b] = conv_w @ relu(bn(out_r[b]))  (WMMA f32) --------
__global__ void k_final(const float* __restrict__ outacc, const float* __restrict__ conv_w,
                        const float* __restrict__ conv_b, const float* __restrict__ gat_bias,
                        const float* __restrict__ scale, const float* __restrict__ shift,
                        float* __restrict__ y) {
    int wave = blockIdx.x * (blockDim.x >> 5) + (threadIdx.x >> 5);
    int lane = threadIdx.x & 31;
    int ltile = wave & 1023;                 // 16384/16 = 1024 column tiles
    int it    = (wave >> 10) & 3;            // 64/16 = 4 row tiles
    int b     = wave >> 12;                  // 4 batches
    int i0 = it * 16, l0 = ltile * 16;
    int half = lane >> 4, l16 = lane & 15;
    int col_l = l0 + l16;                    // sequence position (B/D column)
    int j = col_l & 63;                      // BN feature for this flat column
    float sc = scale[j], sh = shift[j], gb = gat_bias[j];
    const float* aptr  = conv_w + (size_t)(i0 + l16) * C_OUT + half * 2;  // conv_w[i][o]
    const float* bbase = outacc + (size_t)b * (C_OUT * LSEQ) + col_l;     // + o*LSEQ
    v8f c = {};
#pragma unroll
    for (int k0 = 0; k0 < C_OUT; k0 += 4) {
        v2f a, bb;
        a.x = aptr[k0 + 0]; a.y = aptr[k0 + 1];
        float v0 = bbase[(size_t)(k0 + half * 2 + 0) * LSEQ];
        float v1 = bbase[(size_t)(k0 + half * 2 + 1) * LSEQ];
        bb.x = fmaxf(0.f, (v0 + gb) * sc + sh);
        bb.y = fmaxf(0.f, (v1 + gb) * sc + sh);
        c = __builtin_amdgcn_wmma_f32_16x16x4_f32(false, a, false, bb, (short)0, c, false, false);
    }
#pragma unroll
    for (int r = 0; r < 8; ++r) {
        int i = i0 + half * 8 + r;
        y[(size_t)b * (C_OUT * LSEQ) + (size_t)i * LSEQ + col_l] = c[r] + conv_b[i];
    }
}

// ---------------- host side ----------------
extern "C" void kernel_launch(void* const* d_in, const int* in_sizes, int n_in,
                              void* d_out, int out_size, void* d_ws, size_t ws_size,
                              hipStream_t stream) {
    const float* x        = (const float*)d_in[0];   // [4,64,16384] -> xf [65536,64]
    const int*   ei       = (const int*)  d_in[1];   // [2, E]
    const float* W        = (const float*)d_in[2];   // [192, 64]
    const float* att_src  = (const float*)d_in[3];   // [3, 64]
    const float* att_dst  = (const float*)d_in[4];   // [3, 64]
    const float* gat_bias = (const float*)d_in[5];   // [64]
    const float* bn_gamma = (const float*)d_in[6];   // [64]
    const float* bn_beta  = (const float*)d_in[7];   // [64]
    const float* conv_w   = (const float*)d_in[8];   // [64, 64]
    const float* conv_b   = (const float*)d_in[9];   // [64]
    float* y = (float*)d_out;                        // [4, 64, 16384]

    // workspace layout (floats)
    float* w = (float*)d_ws;
    float* h      = w;                       w += (size_t)NNODE * HF;        // 12,582,912
    float* a_s    = w;                       w += (size_t)NNODE * NHEAD;
    float* a_d    = w;                       w += (size_t)NNODE * NHEAD;
    float* m      = w;                       w += (size_t)NNODE * NHEAD;
    float* denom  = w;                       w += (size_t)NNODE * NHEAD;
    float* exbuf  = w;                       w += (size_t)ETOT  * NHEAD;     // 3,342,336
    float* outacc = w;                       w += (size_t)NNODE * C_OUT;     // 4,194,304
    float* stats  = w;                       w += 128;                       // sum | sumsq
    float* scale  = w;                       w += 64;
    float* shift  = w;                       w += 64;

    // K0: init m=-inf, denom=0, outacc=0, stats=0
    k_init<<<(NNODE * C_OUT) / 256, 256, 0, stream>>>(m, denom, outacc, stats);

    // K1: h = xf @ W^T   (4096 x 12 tiles, 4 waves/block)
    {
        int waves = (NNODE / 16) * (HF / 16);            // 49152
        k_gemm_h<<<waves / 4, 128, 0, stream>>>(x, W, h);
    }

    // K2: attention dots
    k_att<<<(NNODE * NHEAD + 255) / 256, 256, 0, stream>>>(h, att_src, att_dst, a_s, a_d);

    // K3: segment max
    k_segmax<<<(ETOT * NHEAD + 255) / 256, 256, 0, stream>>>(ei, a_s, a_d, m);

    // K4: exp + segment sum
    k_expsum<<<(ETOT * NHEAD + 255) / 256, 256, 0, stream>>>(ei, a_s, a_d, m, exbuf, denom);

    // K5: weighted scatter, wave per edge (8 waves / 256-thread block)
    k_scatter<<<(ETOT + 7) / 8, 256, 0, stream>>>(ei, h, exbuf, denom, outacc);

    // K6/K7: batch-norm statistics and folded params
    k_stats<<<256, 256, 0, stream>>>(outacc, gat_bias, stats);
    k_bnparams<<<1, 64, 0, stream>>>(stats, bn_gamma, bn_beta, scale, shift);

    // K8: fused BN+ReLU + 1x1 conv GEMM  (4 b x 4 itile x 1024 ltile waves)
    {
        int waves = NB * 4 * 1024;                       // 16384
        k_final<<<waves / 4, 128, 0, stream>>>(outacc, conv_w, conv_b, gat_bias, scale, shift, y);
    }
}